// CHARM_83940840833097
// MI455X (gfx1250) — compile-verified
//
#include <hip/hip_runtime.h>

// ---------------------------------------------------------------------------
// CHARM GNN forward for MI455X (gfx1250).
// f32 WMMA (V_WMMA_F32_16X16X4_F32) for all GEMMs; CDNA5 async-to-LDS
// double-buffered gather for the edge-message pipeline.
// N=10000 nodes, E=320000 edges, HID=128, EDGE_D=16, 2 layers.
// ---------------------------------------------------------------------------

typedef __attribute__((ext_vector_type(2))) float v2f;
typedef __attribute__((ext_vector_type(4))) float v4f;
typedef __attribute__((ext_vector_type(8))) float v8f;

#define NN        10000
#define EE        320000
#define HID       128
#define EDGE_D    16
#define MSG_K     146          // HID + EDGE_D + 2
#define MSG_KP    148          // padded to multiple of 4
#define AM_STRIDE 148          // LDS stride for A tile (bank-friendly, 16B mult)
#define T_STRIDE  132          // LDS stride for intermediate tile
#define UPD_K     256
#define UPD_KP    260

static __device__ __forceinline__ v8f wmma4(v2f a, v2f b, v8f c) {
  // D = A(16x4) * B(4x16) + C(16x16), all f32.
  return __builtin_amdgcn_wmma_f32_16x16x4_f32(false, a, false, b, (short)0, c,
                                               false, false);
}

// ---------------------------------------------------------------------------
// CDNA5 async global->LDS helpers (ASYNCcnt-tracked, no VGPR round trip).
// Builtin signature (from hipcc diagnostics): param0 = int __vector(N)
// pointer in the global (__device__) address space, param1 = same in LDS
// (__shared__) address space, then imm offset and cpol.
// Fallback: plain vector copies (compiler handles dscnt at barriers).
// ---------------------------------------------------------------------------
#if defined(__has_builtin)
#if __has_builtin(__builtin_amdgcn_global_load_async_to_lds_b128) && \
    __has_builtin(__builtin_amdgcn_global_load_async_to_lds_b64)
#define USE_ASYNC_LDS 1
#endif
#endif
#ifndef USE_ASYNC_LDS
#define USE_ASYNC_LDS 0
#endif

#if USE_ASYNC_LDS
typedef int vec4i __attribute__((__vector_size__(16)));
typedef int vec2i __attribute__((__vector_size__(8)));
typedef __attribute__((address_space(1))) vec4i* g128_t;
typedef __attribute__((address_space(3))) vec4i* l128_t;
typedef __attribute__((address_space(1))) vec2i* g64_t;
typedef __attribute__((address_space(3))) vec2i* l64_t;
static __device__ __forceinline__ void copy_b128(const void* g, void* l) {
  __builtin_amdgcn_global_load_async_to_lds_b128((g128_t)g, (l128_t)l, 0, 0);
}
static __device__ __forceinline__ void copy_b64(const void* g, void* l) {
  __builtin_amdgcn_global_load_async_to_lds_b64((g64_t)g, (l64_t)l, 0, 0);
}
#else
static __device__ __forceinline__ void copy_b128(const void* g, void* l) {
  *(v4f*)l = *(const v4f*)g;
}
static __device__ __forceinline__ void copy_b64(const void* g, void* l) {
  *(v2f*)l = *(const v2f*)g;
}
#endif

template <int N>
static __device__ __forceinline__ void wait_async() {
#if USE_ASYNC_LDS
#if __has_builtin(__builtin_amdgcn_s_wait_asynccnt)
  __builtin_amdgcn_s_wait_asynccnt((short)N);
#else
  asm volatile("s_wait_asynccnt %0" ::"i"(N));
#endif
#endif
}

// ---------------------------------------------------------------------------
// Generic Y[16-row tile] = act(X @ W + b).  One block = 16 rows, wave w owns
// 16 output columns.  K is compile-time.  dout <= 128.
// ---------------------------------------------------------------------------
template <int K, bool RELU>
__global__ __launch_bounds__(256) void gemm_node_kernel(
    const float* __restrict__ X, const float* __restrict__ W,
    const float* __restrict__ bias, float* __restrict__ Y, int dout) {
  constexpr int KP = K + 4;
  __shared__ float Xs[16 * KP];

  const int n0  = blockIdx.x * 16;
  const int tid = threadIdx.x;

  for (int t = tid; t < 16 * K; t += 256) {
    int row = t / K, c = t % K;
    Xs[row * KP + c] = X[(n0 + row) * K + c];
  }
  __syncthreads();

  const int wave = tid >> 5;
  const int lane = tid & 31;
  const int half = lane >> 4;   // 0: K even pair, 1: K odd pair
  const int mn   = lane & 15;   // M for A, N for B/C/D
  const int col0 = wave * 16;
  if (col0 >= dout) return;

  v8f acc = {};
#pragma unroll
  for (int k = 0; k < K; k += 4) {
    const int ka = k + 2 * half;
    v2f a, b;
    a.x = Xs[mn * KP + ka];
    a.y = Xs[mn * KP + ka + 1];
    b.x = W[ka * dout + col0 + mn];
    b.y = W[(ka + 1) * dout + col0 + mn];
    acc = wmma4(a, b, acc);
  }

  const float bb = bias[col0 + mn];
#pragma unroll
  for (int v = 0; v < 8; ++v) {
    const int m = v + 8 * half;
    float r = acc[v] + bb;
    if (RELU) r = fmaxf(r, 0.0f);
    Y[(n0 + m) * dout + col0 + mn] = r;
  }
}

// ---------------------------------------------------------------------------
// Edge message MLP + scatter-add.  Persistent blocks; each wave caches its
// 16-column slice of W1 (146x128) and W2 (128x128) in registers.  The
// 16-edge A tiles are gathered memory->LDS with CDNA5 async loads, double
// buffered so tile t+1's gather overlaps tile t's WMMA work.
// Every thread issues exactly 4 async ops per tile (3x b128 + 1x b64), so
// s_wait_asynccnt<=4 after issuing the next batch drains exactly the
// previous one (async loads complete in order).
// ---------------------------------------------------------------------------
__global__ __launch_bounds__(256) void edge_msg_kernel(
    const float* __restrict__ h, const int* __restrict__ ei,
    const float* __restrict__ eattr, const float* __restrict__ emark,
    const float* __restrict__ W1, const float* __restrict__ b1,
    const float* __restrict__ W2, const float* __restrict__ b2,
    float* __restrict__ aggr, int ntiles) {
  __shared__ __align__(16) float Am[2][16 * AM_STRIDE];  // double-buffered A
  __shared__ __align__(16) float T1[16 * T_STRIDE];      // relu(msg1) tile
  __shared__ __align__(16) float dump[4];                // async overflow sink

  const int tid  = threadIdx.x;
  const int wave = tid >> 5;
  const int lane = tid & 31;
  const int half = lane >> 4;
  const int mn   = lane & 15;
  const int col0 = wave * 16;
  const int col  = col0 + mn;

  // Register-resident B fragments for this wave's 16-column weight slices.
  v2f B1f[MSG_KP / 4];  // 37
  v2f B2f[HID / 4];     // 32
#pragma unroll
  for (int i = 0; i < MSG_KP / 4; ++i) {
    const int ka = 4 * i + 2 * half;
    B1f[i].x = (ka     < MSG_K) ? W1[ka * HID + col]       : 0.0f;
    B1f[i].y = (ka + 1 < MSG_K) ? W1[(ka + 1) * HID + col] : 0.0f;
  }
#pragma unroll
  for (int i = 0; i < HID / 4; ++i) {
    const int ka = 4 * i + 2 * half;
    B2f[i].x = W2[ka * HID + col];
    B2f[i].y = W2[(ka + 1) * HID + col];
  }
  const float bias1 = b1[col];
  const float bias2 = b2[col];

  // Pre-zero pad columns 146..147 of both buffers (async gather never
  // touches them).
  if (tid < 64) {
    const int b = tid >> 5, r = tid & 15, c = (tid >> 4) & 1;
    Am[b][r * AM_STRIDE + MSG_K + c] = 0.0f;
  }
  __syncthreads();

  // Issue the gather for one 16-edge tile into AmBuf.
  // b128 domain: 768 slots (3/thread); slots 0..575 = 16 rows x 36 float4
  // (32 from h[src], 4 from edge_attr); rest hit the dump sink.
  // b64: one per thread; threads 0..15 fetch edge_mark rows.
  auto issue_gather = [&](float* AmBuf, int e0) {
#pragma unroll
    for (int it = 0; it < 3; ++it) {
      const int s = tid + it * 256;
      const float* g;
      float* l;
      if (s < 576) {
        const int r = s / 36;
        const int j = s - r * 36;
        const int e = e0 + r;
        if (j < 32) {
          g = &h[(size_t)ei[e] * HID + 4 * j];          // src = ei[0][e]
        } else {
          g = &eattr[(size_t)e * EDGE_D + 4 * (j - 32)];
        }
        l = &AmBuf[r * AM_STRIDE + 4 * j];
      } else {
        g = h;
        l = dump;
      }
      copy_b128(g, l);
    }
    {
      const float* g;
      float* l;
      if (tid < 16) {
        g = &emark[(size_t)(e0 + tid) * 2];
        l = &AmBuf[tid * AM_STRIDE + (HID + EDGE_D)];
      } else {
        g = emark;
        l = dump;
      }
      copy_b64(g, l);
    }
  };

  int tile = blockIdx.x;
  int buf  = 0;
  if (tile < ntiles) issue_gather(Am[0], tile * 16);  // prologue

  for (; tile < ntiles; tile += gridDim.x) {
    const int next = tile + gridDim.x;
    if (next < ntiles) {
      issue_gather(Am[buf ^ 1], next * 16);  // overlap next gather
      wait_async<4>();                       // drain current tile's batch
    } else {
      wait_async<0>();
    }
    __syncthreads();  // Am[buf] resident for all waves

    const float* A  = Am[buf];
    const int    e0 = tile * 16;

    // Stage 1: T1 = relu(A @ W1 + b1)
    v8f acc = {};
#pragma unroll
    for (int i = 0; i < MSG_KP / 4; ++i) {
      const int ka = 4 * i + 2 * half;
      v2f a;
      a.x = A[mn * AM_STRIDE + ka];
      a.y = A[mn * AM_STRIDE + ka + 1];
      acc = wmma4(a, B1f[i], acc);
    }
#pragma unroll
    for (int v = 0; v < 8; ++v) {
      const int m = v + 8 * half;
      T1[m * T_STRIDE + col] = fmaxf(acc[v] + bias1, 0.0f);
    }
    __syncthreads();

    // Stage 2: m = T1 @ W2 + b2, scatter-add into aggr[dst].
    v8f acc2 = {};
#pragma unroll
    for (int i = 0; i < HID / 4; ++i) {
      const int ka = 4 * i + 2 * half;
      v2f a;
      a.x = T1[mn * T_STRIDE + ka];
      a.y = T1[mn * T_STRIDE + ka + 1];
      acc2 = wmma4(a, B2f[i], acc2);
    }
#pragma unroll
    for (int v = 0; v < 8; ++v) {
      const int m = v + 8 * half;
      const int d = ei[EE + (e0 + m)];  // dst = ei[1][e]
      atomicAdd(&aggr[d * HID + col], acc2[v] + bias2);
    }
    __syncthreads();  // protect Am/T1 reuse next tile
    buf ^= 1;
  }
}

// ---------------------------------------------------------------------------
// Node update: h = relu(h + up2(relu(up1([h | aggr/deg]))))   (in place)
// ---------------------------------------------------------------------------
__global__ __launch_bounds__(256) void update_kernel(
    float* __restrict__ h, const float* __restrict__ aggr,
    const float* __restrict__ deg, const float* __restrict__ W1,
    const float* __restrict__ b1, const float* __restrict__ W2,
    const float* __restrict__ b2) {
  __shared__ float U[16 * UPD_KP];     // [16 nodes][256]
  __shared__ float T[16 * T_STRIDE];   // [16 nodes][128]

  const int n0  = blockIdx.x * 16;
  const int tid = threadIdx.x;

  for (int t = tid; t < 16 * UPD_K; t += 256) {
    const int row = t / UPD_K, c = t % UPD_K;
    const int n = n0 + row;
    float v;
    if (c < HID) {
      v = h[n * HID + c];
    } else {
      float d = deg[n];
      d = (d == 0.0f) ? 1.0f : d;
      v = aggr[n * HID + (c - HID)] / d;
    }
    U[row * UPD_KP + c] = v;
  }
  __syncthreads();

  const int wave = tid >> 5;
  const int lane = tid & 31;
  const int half = lane >> 4;
  const int mn   = lane & 15;
  const int col0 = wave * 16;
  const int col  = col0 + mn;

  v8f acc = {};
#pragma unroll
  for (int k = 0; k < UPD_K; k += 4) {
    const int ka = k + 2 * half;
    v2f a, b;
    a.x = U[mn * UPD_KP + ka];
    a.y = U[mn * UPD_KP + ka + 1];
    b.x = W1[ka * HID + col];
    b.y = W1[(ka + 1) * HID + col];
    acc = wmma4(a, b, acc);
  }
  const float bias1 = b1[col];
#pragma unroll
  for (int v = 0; v < 8; ++v) {
    const int m = v + 8 * half;
    T[m * T_STRIDE + col] = fmaxf(acc[v] + bias1, 0.0f);
  }
  __syncthreads();

  v8f acc2 = {};
#pragma unroll
  for (int k = 0; k < HID; k += 4) {
    const int ka = k + 2 * half;
    v2f a, b;
    a.x = T[mn * T_STRIDE + ka];
    a.y = T[mn * T_STRIDE + ka + 1];
    b.x = W2[ka * HID + col];
    b.y = W2[(ka + 1) * HID + col];
    acc2 = wmma4(a, b, acc2);
  }
  const float bias2 = b2[col];
#pragma unroll
  for (int v = 0; v < 8; ++v) {
    const int m = v + 8 * half;
    const float hold = U[m * UPD_KP + col];  // old h (cols 0..127 of U)
    h[(n0 + m) * HID + col] = fmaxf(hold + acc2[v] + bias2, 0.0f);
  }
}

// ---------------------------------------------------------------------------
// Final scalar head: out[n] = b + dot(z[n, 0:64], w)
// ---------------------------------------------------------------------------
__global__ __launch_bounds__(256) void pred2_kernel(
    const float* __restrict__ z, const float* __restrict__ w,
    const float* __restrict__ b, float* __restrict__ out, int n) {
  const int i = blockIdx.x * blockDim.x + threadIdx.x;
  if (i < n) {
    float acc = b[0];
#pragma unroll
    for (int k = 0; k < 64; ++k) acc += z[i * 64 + k] * w[k];
    out[i] = acc;
  }
}

// ---------------------------------------------------------------------------
// Host launcher.  d_in order (JAX pytree: dict keys sorted, so b before w):
//  0:x 1:edge_index 2:edge_attr 3:edge_mark 4:deg_out
//  5:in_proj.b 6:in_proj.w
//  7+8l..14+8l: layer l {msg1.b, msg1.w, msg2.b, msg2.w, up1.b, up1.w, up2.b, up2.w}
//  23:pred1.b 24:pred1.w 25:pred2.b 26:pred2.w
// ---------------------------------------------------------------------------
extern "C" void kernel_launch(void* const* d_in, const int* in_sizes, int n_in,
                              void* d_out, int out_size, void* d_ws,
                              size_t ws_size, hipStream_t stream) {
  const float* x      = (const float*)d_in[0];
  const int*   ei     = (const int*)d_in[1];
  const float* eattr  = (const float*)d_in[2];
  const float* emark  = (const float*)d_in[3];
  const float* deg    = (const float*)d_in[4];
  const float* inp_b  = (const float*)d_in[5];
  const float* inp_w  = (const float*)d_in[6];
  const float* p1_b   = (const float*)d_in[23];
  const float* p1_w   = (const float*)d_in[24];
  const float* p2_b   = (const float*)d_in[25];
  const float* p2_w   = (const float*)d_in[26];

  float* h    = (float*)d_ws;             // [N,128]
  float* aggr = h + (size_t)NN * HID;     // [N,128]
  float* z    = aggr + (size_t)NN * HID;  // [N,64]

  // h = relu(x @ Win + bin)
  gemm_node_kernel<HID, true><<<NN / 16, 256, 0, stream>>>(x, inp_w, inp_b, h, HID);

  const int ntiles = EE / 16;  // 20000
  for (int l = 0; l < 2; ++l) {
    const float* m1b = (const float*)d_in[7 + 8 * l + 0];
    const float* m1w = (const float*)d_in[7 + 8 * l + 1];
    const float* m2b = (const float*)d_in[7 + 8 * l + 2];
    const float* m2w = (const float*)d_in[7 + 8 * l + 3];
    const float* u1b = (const float*)d_in[7 + 8 * l + 4];
    const float* u1w = (const float*)d_in[7 + 8 * l + 5];
    const float* u2b = (const float*)d_in[7 + 8 * l + 6];
    const float* u2w = (const float*)d_in[7 + 8 * l + 7];

    (void)hipMemsetAsync(aggr, 0, (size_t)NN * HID * sizeof(float), stream);
    edge_msg_kernel<<<2000, 256, 0, stream>>>(h, ei, eattr, emark, m1w, m1b,
                                              m2w, m2b, aggr, ntiles);
    update_kernel<<<NN / 16, 256, 0, stream>>>(h, aggr, deg, u1w, u1b, u2w, u2b);
  }

  // z = relu(h @ Wp1 + bp1), out = z @ wp2 + bp2
  gemm_node_kernel<HID, true><<<NN / 16, 256, 0, stream>>>(h, p1_w, p1_b, z, 64);
  pred2_kernel<<<(NN + 255) / 256, 256, 0, stream>>>(z, p2_w, p2_b,
                                                     (float*)d_out, NN);
}